// SEGNN_53188874993854
// MI455X (gfx1250) — compile-verified
//
// SEGNN (l_max=1 steerable GNN) on MI455X / gfx1250.
//
// Strategy: all heavy math is tall-skinny f32 GEMM over groups of 16 edges /
// nodes -> map each group to one wave32 and use V_WMMA_F32_16X16X4_F32.
// Working set (~8MB) is L2-resident, so the kernel is compute-bound; f32 WMMA
// keeps numerics faithful to the f32 reference while using the matrix pipe.
// Tensor-product features are staged in LDS with a padded row stride (68
// floats) so WMMA A-fragment loads are bank-conflict-free. Message
// aggregation uses global_atomic_add_f32 (pre-scaled by 1/(E-1)).

#include <hip/hip_runtime.h>
#include <cmath>

namespace {
constexpr int   NN   = 16384;               // nodes
constexpr int   EE   = 262144;              // edges
constexpr int   LDA  = 68;                  // padded LDS row stride (floats)
constexpr float NORM = 1.0f / 262143.0f;    // 1/(E-1)
}

typedef float v2f __attribute__((ext_vector_type(2)));
typedef float v8f __attribute__((ext_vector_type(8)));

__device__ __forceinline__ v8f wmma4(v2f a, v2f b, v8f c) {
  // D = A(16x4) * B(4x16) + C(16x16), all f32
  return __builtin_amdgcn_wmma_f32_16x16x4_f32(false, a, false, b, (short)0, c,
                                               false, false);
}

// A: LDS tile, row-major [16][LDA]; W: LDS weights, row-major [K][NC].
// Accumulates A(16xK) * W[:, n0:n0+16] into acc.
// A-fragment layout (ISA 7.12.2, 32-bit A 16x4): lanes 0-15 hold K=k0,k0+1;
// lanes 16-31 hold K=k0+2,k0+3; M = lane&15. B symmetric with N = lane&15.
template <int K>
__device__ __forceinline__ v8f gemm_tile(const float* A, const float* W,
                                         int NC, int n0, int lane, v8f acc) {
  const int m  = lane & 15;
  const int kh = (lane >> 4) << 1;  // 0 or 2
#pragma unroll
  for (int k0 = 0; k0 < K; k0 += 4) {
    v2f a, b;
    a.x = A[m * LDA + k0 + kh];
    a.y = A[m * LDA + k0 + kh + 1];
    b.x = W[(k0 + kh) * NC + n0 + m];
    b.y = W[(k0 + kh + 1) * NC + n0 + m];
    acc = wmma4(a, b, acc);
  }
  return acc;
}

// ---------------------------------------------------------------------------
// Embed: (16s,1v) x attr(1s,1v) -> tp(17s,17v) -> linear -> (16s,16v). Tiny
// K=17 GEMV; plain VALU per node.
// ---------------------------------------------------------------------------
__global__ __launch_bounds__(128) void embed_kernel(
    const float* __restrict__ ns, const float* __restrict__ nv,
    const float* __restrict__ nas, const float* __restrict__ nav,
    const float* __restrict__ Ws, const float* __restrict__ Wv,
    const float* __restrict__ b, float* __restrict__ s_out,
    float* __restrict__ v_out) {
  const int n = blockIdx.x * blockDim.x + threadIdx.x;
  const float as  = nas[n];
  const float av0 = nav[n * 3 + 0], av1 = nav[n * 3 + 1], av2 = nav[n * 3 + 2];
  const float xv0 = nv[n * 3 + 0], xv1 = nv[n * 3 + 1], xv2 = nv[n * 3 + 2];
  float xs[16];
#pragma unroll
  for (int i = 0; i < 16; ++i) xs[i] = ns[n * 16 + i];

  float accS[16];
  float accV[16][3];
#pragma unroll
  for (int o = 0; o < 16; ++o) {
    accS[o] = b[o];
    accV[o][0] = 0.f; accV[o][1] = 0.f; accV[o][2] = 0.f;
  }
#pragma unroll
  for (int k = 0; k < 17; ++k) {
    float tsk, t0, t1, t2;
    if (k < 16) {  // ss = xs*attr_s ; sv = xs*attr_v
      tsk = xs[k] * as;
      t0 = xs[k] * av0; t1 = xs[k] * av1; t2 = xs[k] * av2;
    } else {       // vv = dot(xv, attr_v) ; vs = xv*attr_s
      tsk = xv0 * av0 + xv1 * av1 + xv2 * av2;
      t0 = xv0 * as; t1 = xv1 * as; t2 = xv2 * as;
    }
#pragma unroll
    for (int o = 0; o < 16; ++o) {
      accS[o] += tsk * Ws[k * 16 + o];
      const float w = Wv[k * 16 + o];
      accV[o][0] += t0 * w; accV[o][1] += t1 * w; accV[o][2] += t2 * w;
    }
  }
#pragma unroll
  for (int o = 0; o < 16; ++o) {
    s_out[n * 16 + o] = accS[o];
    v_out[(n * 16 + o) * 3 + 0] = accV[o][0];
    v_out[(n * 16 + o) * 3 + 1] = accV[o][1];
    v_out[(n * 16 + o) * 3 + 2] = accV[o][2];
  }
}

__global__ void zero_kernel(float* __restrict__ p, int n) {
  const int i = blockIdx.x * blockDim.x + threadIdx.x;
  if (i < n) p[i] = 0.0f;
}

// ---------------------------------------------------------------------------
// Edge kernel: per wave = 16 edges. Gather -> tp -> WMMA(gated) -> tp ->
// WMMA -> atomic scatter to receiver aggregates. Block = 64 thr (2 waves).
// ---------------------------------------------------------------------------
__global__ __launch_bounds__(64) void edge_kernel(
    const float* __restrict__ s_cur, const float* __restrict__ v_cur,
    const float* __restrict__ eas_g, const float* __restrict__ eav_g,
    const int* __restrict__ senders, const int* __restrict__ receivers,
    const float* __restrict__ We1s, const float* __restrict__ We1v,
    const float* __restrict__ be1, const float* __restrict__ We2s,
    const float* __restrict__ We2v, const float* __restrict__ be2,
    float* __restrict__ agg_s, float* __restrict__ agg_v) {
  __shared__ float W1s[64 * 32], W1v[64 * 16], W2s[32 * 16], W2v[32 * 16];
  __shared__ float B1[32], B2[16];
  __shared__ float TS[2][16 * LDA];
  __shared__ float TVx[2][16 * LDA], TVy[2][16 * LDA], TVz[2][16 * LDA];
  __shared__ float EAs[2][16], EAv[2][16][3];
  __shared__ int   RCV[2][16];

  const int tid  = threadIdx.x;
  const int w    = tid >> 5;
  const int lane = tid & 31;

  // cooperative weight staging (shared by both waves)
  for (int i = tid; i < 64 * 32; i += 64) W1s[i] = We1s[i];
  for (int i = tid; i < 64 * 16; i += 64) W1v[i] = We1v[i];
  for (int i = tid; i < 32 * 16; i += 64) W2s[i] = We2s[i];
  for (int i = tid; i < 32 * 16; i += 64) W2v[i] = We2v[i];
  if (tid < 32) B1[tid] = be1[tid];
  if (tid < 16) B2[tid] = be2[tid];

  const int ebase = blockIdx.x * 32 + w * 16;
  const int r = lane & 15;   // edge row within tile
  const int h = lane >> 4;   // 0: sender channels, 1: receiver channels
  const int e = ebase + r;

  const int   sd  = senders[e];
  const int   rc  = receivers[e];
  const float ea  = eas_g[e];
  const float ev0 = eav_g[e * 3 + 0];
  const float ev1 = eav_g[e * 3 + 1];
  const float ev2 = eav_g[e * 3 + 2];
  if (h == 0) {
    RCV[w][r] = rc;
    EAs[w][r] = ea;
    EAv[w][r][0] = ev0; EAv[w][r][1] = ev1; EAv[w][r][2] = ev2;
  }
  const int src = h ? rc : sd;

  float* ts  = TS[w];
  float* tv0 = TVx[w];
  float* tv1 = TVy[w];
  float* tv2 = TVz[w];

  // tp #1: ms(32)=s[sd]++s[rcv], mv(32)=v[sd]++v[rcv] against edge attrs.
  // Scalars: [ms*ea_s (0..31), dot(mv,ea_v) (32..63)]
  // Vectors: [ms*ea_v (0..31), mv*ea_s (32..63)]
#pragma unroll
  for (int c = 0; c < 16; ++c) {
    const int   ch  = h * 16 + c;
    const float msv = s_cur[src * 16 + c];
    const float mx  = v_cur[(src * 16 + c) * 3 + 0];
    const float my  = v_cur[(src * 16 + c) * 3 + 1];
    const float mz  = v_cur[(src * 16 + c) * 3 + 2];
    ts[r * LDA + ch]      = msv * ea;
    ts[r * LDA + 32 + ch] = mx * ev0 + my * ev1 + mz * ev2;
    tv0[r * LDA + ch] = msv * ev0;
    tv1[r * LDA + ch] = msv * ev1;
    tv2[r * LDA + ch] = msv * ev2;
    tv0[r * LDA + 32 + ch] = mx * ea;
    tv1[r * LDA + 32 + ch] = my * ea;
    tv2[r * LDA + 32 + ch] = mz * ea;
  }
  __syncthreads();

  // GEMM stage 1: ls[16x32] = ts @ We1_s ; lv_a[16x16] = tv_a @ We1_v
  v8f aS0 = {0, 0, 0, 0, 0, 0, 0, 0};
  v8f aS1 = aS0, aV0 = aS0, aV1 = aS0, aV2 = aS0;
  aS0 = gemm_tile<64>(ts, W1s, 32, 0, lane, aS0);
  aS1 = gemm_tile<64>(ts, W1s, 32, 16, lane, aS1);
  aV0 = gemm_tile<64>(tv0, W1v, 16, 0, lane, aV0);
  aV1 = gemm_tile<64>(tv1, W1v, 16, 0, lane, aV1);
  aV2 = gemm_tile<64>(tv2, W1v, 16, 0, lane, aV2);

  // Gate: feat = silu(ls[:, :16]+b), gate = sigmoid(ls[:, 16:]+b).
  // D-tile element (vgpr j, lane): row = j + 8*(lane>>4), col = lane&15 —
  // identical layout for scalar and vector tiles, so gating is elementwise.
  const int   c0 = lane & 15;
  const float bF = B1[c0], bG = B1[16 + c0];
  float feat[8], gx[8], gy[8], gz[8];
#pragma unroll
  for (int j = 0; j < 8; ++j) {
    const float x = aS0[j] + bF;
    feat[j] = x / (1.0f + expf(-x));                  // silu
    const float g = 1.0f / (1.0f + expf(-(aS1[j] + bG)));
    gx[j] = aV0[j] * g; gy[j] = aV1[j] * g; gz[j] = aV2[j] * g;
  }
  __syncthreads();

  // tp #2 (16s,16v) x edge attrs -> (32s,32v), transposed back into LDS
#pragma unroll
  for (int j = 0; j < 8; ++j) {
    const int   rr = j + 8 * h;
    const float as = EAs[w][rr];
    const float a0 = EAv[w][rr][0], a1 = EAv[w][rr][1], a2 = EAv[w][rr][2];
    ts[rr * LDA + c0]      = feat[j] * as;
    ts[rr * LDA + 16 + c0] = gx[j] * a0 + gy[j] * a1 + gz[j] * a2;
    tv0[rr * LDA + c0] = feat[j] * a0; tv0[rr * LDA + 16 + c0] = gx[j] * as;
    tv1[rr * LDA + c0] = feat[j] * a1; tv1[rr * LDA + 16 + c0] = gy[j] * as;
    tv2[rr * LDA + c0] = feat[j] * a2; tv2[rr * LDA + 16 + c0] = gz[j] * as;
  }
  __syncthreads();

  // GEMM stage 2 (K=32)
  v8f bS = {0, 0, 0, 0, 0, 0, 0, 0};
  v8f bV0 = bS, bV1 = bS, bV2 = bS;
  bS  = gemm_tile<32>(ts, W2s, 16, 0, lane, bS);
  bV0 = gemm_tile<32>(tv0, W2v, 16, 0, lane, bV0);
  bV1 = gemm_tile<32>(tv1, W2v, 16, 0, lane, bV1);
  bV2 = gemm_tile<32>(tv2, W2v, 16, 0, lane, bV2);

  // Scatter (bias added per edge exactly as reference; norm pre-applied)
  const float b2 = B2[c0];
#pragma unroll
  for (int j = 0; j < 8; ++j) {
    const int rr   = j + 8 * h;
    const int node = RCV[w][rr];
    atomicAdd(&agg_s[node * 16 + c0], (bS[j] + b2) * NORM);
    atomicAdd(&agg_v[(node * 16 + c0) * 3 + 0], bV0[j] * NORM);
    atomicAdd(&agg_v[(node * 16 + c0) * 3 + 1], bV1[j] * NORM);
    atomicAdd(&agg_v[(node * 16 + c0) * 3 + 2], bV2[j] * NORM);
  }
}

// ---------------------------------------------------------------------------
// Node update: per wave = 16 nodes. cs=concat(agg_s,s), cv=concat(agg_v,v),
// tp with node attrs -> WMMA(K=64) -> residual in place.
// ---------------------------------------------------------------------------
__global__ __launch_bounds__(64) void node_kernel(
    float* __restrict__ s_cur, float* __restrict__ v_cur,
    const float* __restrict__ agg_s, const float* __restrict__ agg_v,
    const float* __restrict__ nas_g, const float* __restrict__ nav_g,
    const float* __restrict__ Wns, const float* __restrict__ Wnv,
    const float* __restrict__ bn) {
  __shared__ float WS[64 * 16], WV[64 * 16], BN[16];
  __shared__ float TS[2][16 * LDA];
  __shared__ float TVx[2][16 * LDA], TVy[2][16 * LDA], TVz[2][16 * LDA];

  const int tid  = threadIdx.x;
  const int w    = tid >> 5;
  const int lane = tid & 31;

  for (int i = tid; i < 64 * 16; i += 64) { WS[i] = Wns[i]; WV[i] = Wnv[i]; }
  if (tid < 16) BN[tid] = bn[tid];

  const int nbase = blockIdx.x * 32 + w * 16;
  const int r = lane & 15;
  const int h = lane >> 4;   // 0: agg channels, 1: state channels
  const int node = nbase + r;

  const float na  = nas_g[node];
  const float n0v = nav_g[node * 3 + 0];
  const float n1v = nav_g[node * 3 + 1];
  const float n2v = nav_g[node * 3 + 2];

  const float* srcS = h ? (s_cur + node * 16) : (agg_s + node * 16);
  const float* srcV = h ? (v_cur + node * 48) : (agg_v + node * 48);

  float* ts  = TS[w];
  float* tv0 = TVx[w];
  float* tv1 = TVy[w];
  float* tv2 = TVz[w];
#pragma unroll
  for (int c = 0; c < 16; ++c) {
    const int   ch = h * 16 + c;
    const float cs = srcS[c];
    const float cx = srcV[c * 3 + 0];
    const float cy = srcV[c * 3 + 1];
    const float cz = srcV[c * 3 + 2];
    ts[r * LDA + ch]      = cs * na;
    ts[r * LDA + 32 + ch] = cx * n0v + cy * n1v + cz * n2v;
    tv0[r * LDA + ch] = cs * n0v;
    tv1[r * LDA + ch] = cs * n1v;
    tv2[r * LDA + ch] = cs * n2v;
    tv0[r * LDA + 32 + ch] = cx * na;
    tv1[r * LDA + 32 + ch] = cy * na;
    tv2[r * LDA + 32 + ch] = cz * na;
  }
  __syncthreads();

  v8f aS = {0, 0, 0, 0, 0, 0, 0, 0};
  v8f aV0 = aS, aV1 = aS, aV2 = aS;
  aS  = gemm_tile<64>(ts, WS, 16, 0, lane, aS);
  aV0 = gemm_tile<64>(tv0, WV, 16, 0, lane, aV0);
  aV1 = gemm_tile<64>(tv1, WV, 16, 0, lane, aV1);
  aV2 = gemm_tile<64>(tv2, WV, 16, 0, lane, aV2);

  const int   c0  = lane & 15;
  const float bnc = BN[c0];
#pragma unroll
  for (int j = 0; j < 8; ++j) {
    const int rr = j + 8 * h;
    const int nd = nbase + rr;
    s_cur[nd * 16 + c0] += aS[j] + bnc;               // residual
    v_cur[(nd * 16 + c0) * 3 + 0] += aV0[j];
    v_cur[(nd * 16 + c0) * 3 + 1] += aV1[j];
    v_cur[(nd * 16 + c0) * 3 + 2] += aV2[j];
  }
}

// ---------------------------------------------------------------------------
extern "C" void kernel_launch(void* const* d_in, const int* in_sizes, int n_in,
                              void* d_out, int out_size, void* d_ws,
                              size_t ws_size, hipStream_t stream) {
  (void)in_sizes; (void)n_in; (void)out_size; (void)ws_size;

  const float* node_s      = (const float*)d_in[0];
  const float* node_v      = (const float*)d_in[1];
  const float* node_attr_s = (const float*)d_in[2];
  const float* node_attr_v = (const float*)d_in[3];
  const float* edge_attr_s = (const float*)d_in[4];
  const float* edge_attr_v = (const float*)d_in[5];
  const int*   senders     = (const int*)d_in[6];
  const int*   receivers   = (const int*)d_in[7];
  const float* W_emb_s = (const float*)d_in[8];
  const float* W_emb_v = (const float*)d_in[9];
  const float* b_emb   = (const float*)d_in[10];
  const float* We1_s   = (const float*)d_in[11];
  const float* We1_v   = (const float*)d_in[12];
  const float* be1     = (const float*)d_in[13];
  const float* We2_s   = (const float*)d_in[14];
  const float* We2_v   = (const float*)d_in[15];
  const float* be2     = (const float*)d_in[16];
  const float* Wn_s    = (const float*)d_in[17];
  const float* Wn_v    = (const float*)d_in[18];
  const float* bn      = (const float*)d_in[19];

  float* ws    = (float*)d_ws;
  float* s_cur = ws;                          // [N,16]
  float* v_cur = s_cur + (size_t)NN * 16;     // [N,16,3]
  float* agg_s = v_cur + (size_t)NN * 48;     // [N,16]
  float* agg_v = agg_s + (size_t)NN * 16;     // [N,16,3]

  embed_kernel<<<NN / 128, 128, 0, stream>>>(node_s, node_v, node_attr_s,
                                             node_attr_v, W_emb_s, W_emb_v,
                                             b_emb, s_cur, v_cur);
  for (int t = 0; t < 3; ++t) {
    zero_kernel<<<(NN * 64) / 256, 256, 0, stream>>>(agg_s, NN * 64);
    edge_kernel<<<EE / 32, 64, 0, stream>>>(
        s_cur, v_cur, edge_attr_s, edge_attr_v, senders, receivers,
        We1_s + t * 64 * 32, We1_v + t * 64 * 16, be1 + t * 32,
        We2_s + t * 32 * 16, We2_v + t * 32 * 16, be2 + t * 16, agg_s, agg_v);
    node_kernel<<<NN / 32, 64, 0, stream>>>(
        s_cur, v_cur, agg_s, agg_v, node_attr_s, node_attr_v,
        Wn_s + t * 64 * 16, Wn_v + t * 64 * 16, bn + t * 16);
  }
  hipMemcpyAsync(d_out, s_cur, (size_t)NN * 16 * sizeof(float),
                 hipMemcpyDeviceToDevice, stream);
  hipMemcpyAsync((float*)d_out + (size_t)NN * 16, v_cur,
                 (size_t)NN * 48 * sizeof(float), hipMemcpyDeviceToDevice,
                 stream);
}